// BinCalibrationContributionLoss_29781303231105
// MI455X (gfx1250) — compile-verified
//
#include <hip/hip_runtime.h>
#include <stdint.h>

#define CDIM   100
#define NBINS  15
#define GAMMA  0.047f
#define ROWB   128          // padded LDS row buffer (floats)
#define P1_BLOCKS 4096
#define P2_BLOCKS 2048

__device__ __forceinline__ float waveSum(float v) {
    #pragma unroll
    for (int o = 16; o > 0; o >>= 1) v += __shfl_xor(v, o, 32);
    return v;
}

__device__ __forceinline__ void asyncRowB128(uint32_t ldsOff, const float* gaddr) {
    // 16 bytes per enabled lane, global -> LDS, tracked with ASYNCcnt
    asm volatile("global_load_async_to_lds_b128 %0, %1, off"
                 :: "v"(ldsOff), "v"(gaddr) : "memory");
}

__global__ void ece_init(float* gbins) {
    int t = threadIdx.x;
    if (t < 3 * NBINS) gbins[t] = 0.0f;
}

__global__ __launch_bounds__(256)
void ece_pass1(const float* __restrict__ x, const int* __restrict__ y,
               float* __restrict__ conf_o, float* __restrict__ picked_o,
               unsigned char* __restrict__ pk_o, float* __restrict__ gbins, int n) {
    __shared__ float srow[8][2][ROWB];     // 8 waves x double buffer
    __shared__ float sbins[3 * NBINS];

    const int tid  = threadIdx.x;
    const int lane = tid & 31;
    const int wid  = tid >> 5;

    if (tid < 3 * NBINS) sbins[tid] = 0.0f;
    __syncthreads();

    float* b0 = &srow[wid][0][0];
    float* b1 = &srow[wid][1][0];
    // flat pointer low 32 bits == LDS byte offset
    const uint32_t l0 = (uint32_t)(uintptr_t)(b0 + lane * 4);
    const uint32_t l1 = (uint32_t)(uintptr_t)(b1 + lane * 4);

    long long row    = (long long)blockIdx.x * 8 + wid;
    const long long stride = (long long)gridDim.x * 8;

    if (row < n && lane < 25)
        asyncRowB128(l0, x + row * CDIM + lane * 4);

    int cur = 0;
    for (; row < n; row += stride) {
        const long long nxt = row + stride;
        if (nxt < n) {
            if (lane < 25)
                asyncRowB128(cur ? l0 : l1, x + nxt * CDIM + lane * 4);
            asm volatile("s_wait_asynccnt 1" ::: "memory");
        } else {
            asm volatile("s_wait_asynccnt 0" ::: "memory");
        }
        const float* buf = cur ? b1 : b0;

        // row max + first-index argmax
        float m = -3.4e38f; int am = 0;
        #pragma unroll
        for (int k = 0; k < 4; ++k) {
            int j = lane + 32 * k;
            if (j < CDIM) { float f = buf[j]; if (f > m) { m = f; am = j; } }
        }
        #pragma unroll
        for (int o = 16; o > 0; o >>= 1) {
            float mo = __shfl_xor(m, o, 32);
            int   ao = __shfl_xor(am, o, 32);
            if (mo > m || (mo == m && ao < am)) { m = mo; am = ao; }
        }

        // sum of exp(x - m)
        float s = 0.0f;
        #pragma unroll
        for (int k = 0; k < 4; ++k) {
            int j = lane + 32 * k;
            if (j < CDIM) s += expf(buf[j] - m);
        }
        s = waveSum(s);

        const int   yl = y[row];
        const float xy = buf[yl];

        if (lane == 0) {
            const float conf   = 1.0f / s;                 // exp(0)/sum
            const float picked = xy - m - logf(s);         // log_softmax at label
            const int   accI   = (am == yl) ? 1 : 0;
            // searchsorted(left) on uppers=(i+1)/15  ==  ceil(conf*15)-1, clamped
            int b = (int)ceilf(conf * (float)NBINS) - 1;
            b = b < 0 ? 0 : (b > NBINS - 1 ? NBINS - 1 : b);

            atomicAdd(&sbins[b],             1.0f);
            atomicAdd(&sbins[NBINS + b],     conf);
            atomicAdd(&sbins[2 * NBINS + b], (float)accI);

            conf_o[row]   = conf;
            picked_o[row] = picked;
            pk_o[row]     = (unsigned char)(b | (accI << 4));
        }
        cur ^= 1;
    }
    __syncthreads();
    if (tid < 3 * NBINS) atomicAdd(&gbins[tid], sbins[tid]);
}

__global__ __launch_bounds__(256)
void ece_pass2(const float* __restrict__ conf_i, const float* __restrict__ picked_i,
               const unsigned char* __restrict__ pk_i, const float* __restrict__ gbins,
               float* __restrict__ partial, int n) {
    __shared__ float sb[3 * NBINS];
    __shared__ float sw[8];
    const int tid = threadIdx.x;
    if (tid < 3 * NBINS) sb[tid] = gbins[tid];
    __syncthreads();

    float local = 0.0f;
    for (long long i = (long long)blockIdx.x * blockDim.x + tid; i < n;
         i += (long long)gridDim.x * blockDim.x) {
        const unsigned p = pk_i[i];
        const int   b   = p & 15;
        const float acc = (p >> 4) ? 1.0f : 0.0f;
        const float cnt = sb[b];
        const float sc  = sb[NBINS + b];
        const float sa  = sb[2 * NBINS + b];
        const float cf  = conf_i[i];
        const float pk  = picked_i[i];

        const float orig = fabsf(sc - sa) / fmaxf(cnt, 1.0f);
        const float loo  = fabsf((sc - cf) - (sa - acc)) / fmaxf(cnt - 1.0f, 1.0f);
        const float upd  = (cnt > 1.0f) ? loo : 0.0f;
        const float contrib = orig - upd;
        local += -(1.0f + GAMMA * contrib) * pk;
    }

    local = waveSum(local);
    if ((tid & 31) == 0) sw[tid >> 5] = local;
    __syncthreads();
    if (tid == 0) {
        float t = 0.0f;
        #pragma unroll
        for (int i = 0; i < 8; ++i) t += sw[i];
        partial[blockIdx.x] = t;
    }
}

__global__ __launch_bounds__(256)
void ece_finred(const float* __restrict__ partial, int m, float* __restrict__ out, float nf) {
    __shared__ float sw[8];
    const int tid = threadIdx.x;
    float local = 0.0f;
    for (int i = tid; i < m; i += 256) local += partial[i];
    local = waveSum(local);
    if ((tid & 31) == 0) sw[tid >> 5] = local;
    __syncthreads();
    if (tid == 0) {
        float t = 0.0f;
        #pragma unroll
        for (int i = 0; i < 8; ++i) t += sw[i];
        out[0] = t / nf;
    }
}

extern "C" void kernel_launch(void* const* d_in, const int* in_sizes, int n_in,
                              void* d_out, int out_size, void* d_ws, size_t ws_size,
                              hipStream_t stream) {
    const float* x = (const float*)d_in[0];
    const int*   y = (const int*)d_in[1];
    float* out = (float*)d_out;
    const int n = in_sizes[1];              // N rows; in_sizes[0] == N*C

    char* w = (char*)d_ws;
    float*         conf   = (float*)w;
    float*         picked = conf + n;
    unsigned char* packed = (unsigned char*)(picked + n);
    uintptr_t p = (uintptr_t)(packed + n);
    p = (p + 255) & ~(uintptr_t)255;
    float* gbins   = (float*)p;
    float* partial = gbins + 64;

    ece_init  <<<1, 64, 0, stream>>>(gbins);
    ece_pass1 <<<P1_BLOCKS, 256, 0, stream>>>(x, y, conf, picked, packed, gbins, n);
    ece_pass2 <<<P2_BLOCKS, 256, 0, stream>>>(conf, picked, packed, gbins, partial, n);
    ece_finred<<<1, 256, 0, stream>>>(partial, P2_BLOCKS, out, (float)n);
}